// VNN_58574763983266
// MI455X (gfx1250) — compile-verified
//
#include <hip/hip_runtime.h>
#include <math.h>

// ---------------------------------------------------------------------------
// VNN forward for MI455X (gfx1250, wave32).
// FP32 throughout (workload is bandwidth-bound: ~5 GFLOP vs ~1.5 GB scratch
// traffic -> ~65us @ 23.3 TB/s). Matrix work uses V_WMMA_F32_16X16X4_F32.
// ---------------------------------------------------------------------------

#define EPSV   1e-6f
#define BNE    1e-5f
#define TWO_PI 6.2831853071795864769f
#define MAXC   512   // stats stride (>=341)

typedef float v2f __attribute__((ext_vector_type(2)));
typedef float v8f __attribute__((ext_vector_type(8)));

static __device__ __forceinline__ v8f wmma4(v2f a, v2f b, v8f c) {
  // D = A(16x4) * B(4x16) + C(16x16), full fp32
  return __builtin_amdgcn_wmma_f32_16x16x4_f32(false, a, false, b, (short)0, c,
                                               false, false);
}

// ---- time embedding: te[b,o] = GFP(t) @ W^T + bias ------------------------
__global__ void k_te(const float* __restrict__ t, const float* __restrict__ Wf,
                     const float* __restrict__ W, const float* __restrict__ bias,
                     float* __restrict__ out, int Co) {
  int b = blockIdx.x, o = threadIdx.x;
  if (o >= Co) return;
  float tb = t[b];
  float s = bias[o];
  for (int j = 0; j < 64; ++j) {
    float ang = TWO_PI * tb * Wf[j];
    s += sinf(ang) * W[o * 128 + j] + cosf(ang) * W[o * 128 + 64 + j];
  }
  out[b * Co + o] = s;
}

// ---- squared norms per point ----------------------------------------------
__global__ void k_xx(const float* __restrict__ x, float* __restrict__ xx, int total) {
  int i = blockIdx.x * blockDim.x + threadIdx.x;
  if (i >= total) return;
  const float* p = x + (size_t)i * 3;
  xx[i] = p[0] * p[0] + p[1] * p[1] + p[2] * p[2];
}

// ---- kNN: WMMA Gram tiles -> pd chunks in LDS -> register top-10 ----------
__global__ void __launch_bounds__(32)
k_knn(const float* __restrict__ x, const float* __restrict__ xx,
      int* __restrict__ idx, int N) {
  __shared__ float tile[16 * 256];
  __shared__ float xrow[16];

  int lane  = threadIdx.x;
  int laneN = lane & 15;
  bool hi   = lane >= 16;
  int ntr   = N / 16;
  int b     = blockIdx.x / ntr;
  int row0  = (blockIdx.x % ntr) * 16;
  const float* xb = x + (size_t)b * N * 3;

  // A tile: 16 query points, cols = (x,y,z,0). lanes<16 hold K0,K1; >=16 K2,K3.
  v2f a;
  {
    const float* p = xb + (size_t)(row0 + laneN) * 3;
    a.x = hi ? p[2] : p[0];
    a.y = hi ? 0.f  : p[1];
  }
  if (lane < 16) xrow[lane] = xx[b * N + row0 + lane];
  __syncthreads();

  float bv[10]; int bi[10];
#pragma unroll
  for (int q = 0; q < 10; ++q) { bv[q] = -3.0e38f; bi[q] = 0; }

  for (int cb = 0; cb < N / 256; ++cb) {
    int base = cb * 256;
    if (base + 256 < N) __builtin_prefetch(xb + (size_t)(base + 256) * 3, 0, 1);
    for (int tt = 0; tt < 16; ++tt) {
      int col = base + tt * 16 + laneN;
      const float* p = xb + (size_t)col * 3;
      v2f bbv;
      bbv.x = hi ? p[2] : p[0];
      bbv.y = hi ? 0.f  : p[1];
      v8f g = {0.f, 0.f, 0.f, 0.f, 0.f, 0.f, 0.f, 0.f};
      g = wmma4(a, bbv, g);
      float xc = xx[b * N + col];
      int rb = hi ? 8 : 0;
#pragma unroll
      for (int j = 0; j < 8; ++j) {
        int r = rb + j;
        tile[r * 256 + tt * 16 + laneN] = 2.f * g[j] - xrow[r] - xc;
      }
    }
    __syncthreads();
    if (lane < 16) {
      for (int nn = 0; nn < 256; ++nn) {
        float v = tile[lane * 256 + nn];
        if (v > bv[9]) {
          bv[9] = v; bi[9] = base + nn;
#pragma unroll
          for (int q = 9; q > 0; --q) {
            if (bv[q] > bv[q - 1]) {
              float tf = bv[q]; bv[q] = bv[q - 1]; bv[q - 1] = tf;
              int ti = bi[q]; bi[q] = bi[q - 1]; bi[q - 1] = ti;
            }
          }
        }
      }
    }
    __syncthreads();
  }
  if (lane < 16) {
#pragma unroll
    for (int q = 0; q < 10; ++q)
      idx[((size_t)(b * N + row0 + lane)) * 10 + q] = bi[q];
  }
}

// ---- feature build: feat(B,3ch,3,N,k) = [q-p, p, cross(q,p)] ---------------
__global__ void k_feat(const float* __restrict__ x, const int* __restrict__ idx,
                       float* __restrict__ feat, int N, int K, int total) {
  int tid = blockIdx.x * blockDim.x + threadIdx.x;
  if (tid >= total) return;
  int j = tid % K, n = (tid / K) % N, b = tid / (K * N);
  const float* pt = x + ((size_t)b * N + n) * 3;
  int id = idx[(size_t)(b * N + n) * K + j];
  const float* q = x + ((size_t)b * N + id) * 3;
  float px = pt[0], py = pt[1], pz = pt[2];
  float qx = q[0], qy = q[1], qz = q[2];
  float out[9] = { qx - px, qy - py, qz - pz,
                   px, py, pz,
                   qy * pz - qz * py, qz * px - qx * pz, qx * py - qy * px };
  size_t NK = (size_t)N * K;
  size_t basei = (size_t)b * 9 * NK + (size_t)n * K + j;
#pragma unroll
  for (int cd = 0; cd < 9; ++cd) feat[basei + cd * NK] = out[cd];
}

// ---- generic vnlin GEMM: per batch  Y(Co x Mc) = W(Co x Ci) * X(Ci x Mc) ---
// One wave per 16x16 output tile, K-loop by 4 via V_WMMA_F32_16X16X4_F32.
// Optional fused per-(b,row) scale (t_embed applied to p before vnbn).
__global__ void __launch_bounds__(128)
k_gemm(const float* __restrict__ W, const float* __restrict__ X,
       float* __restrict__ Y, const float* __restrict__ te,
       int Bn, int Co, int Ci, int Mc) {
  int lane = threadIdx.x & 31;
  int wid  = blockIdx.x * (blockDim.x >> 5) + (threadIdx.x >> 5);
  int mtiles = (Mc + 15) >> 4;
  int otiles = (Co + 15) >> 4;
  int mt = wid % mtiles;
  int ot = (wid / mtiles) % otiles;
  int b  = wid / (mtiles * otiles);
  if (b >= Bn) return;  // wave-uniform

  int laneN = lane & 15;
  int kb    = (lane >> 4) * 2;          // lanes<16: K0,K1 ; lanes>=16: K2,K3
  int orow  = ot * 16 + laneN;
  int col   = mt * 16 + laneN;
  const float* Xb = X + (size_t)b * Ci * Mc;

  v8f acc = {0.f, 0.f, 0.f, 0.f, 0.f, 0.f, 0.f, 0.f};
  for (int k0 = 0; k0 < Ci; k0 += 4) {
    int ka = k0 + kb;
    v2f av, bvv;
    av.x  = (orow < Co && ka     < Ci) ? W[(size_t)orow * Ci + ka]     : 0.f;
    av.y  = (orow < Co && ka + 1 < Ci) ? W[(size_t)orow * Ci + ka + 1] : 0.f;
    bvv.x = (col < Mc && ka     < Ci) ? Xb[(size_t)ka * Mc + col]       : 0.f;
    bvv.y = (col < Mc && ka + 1 < Ci) ? Xb[(size_t)(ka + 1) * Mc + col] : 0.f;
    acc = wmma4(av, bvv, acc);
  }

  float* Yb = Y + (size_t)b * Co * Mc;
  int rbase = ot * 16 + ((lane >> 4) ? 8 : 0);
#pragma unroll
  for (int j = 0; j < 8; ++j) {
    int row = rbase + j;
    if (row < Co && col < Mc) {
      float v = acc[j];
      if (te) v *= te[b * Co + row];
      Yb[(size_t)row * Mc + col] = v;
    }
  }
}

// ---- misc helpers ----------------------------------------------------------
__global__ void k_zero(float* p, int n) {
  int i = blockIdx.x * blockDim.x + threadIdx.x;
  if (i < n) p[i] = 0.f;
}

// norms + per-channel sum/sumsq.  P is (B,C,3,S); nb is (B,C,S)
__global__ void k_normstats(const float* __restrict__ P, float* __restrict__ nb,
                            float* __restrict__ stats, int C, int S, int Bn) {
  int c = blockIdx.y, tid = threadIdx.x;
  long total = (long)Bn * S;
  float s1 = 0.f, s2 = 0.f;
  for (long m = (long)blockIdx.x * blockDim.x + tid; m < total;
       m += (long)gridDim.x * blockDim.x) {
    long b = m / S, sm = m % S;
    const float* p = P + (((size_t)b * C + c) * 3) * S + sm;
    float v0 = p[0], v1 = p[(size_t)S], v2 = p[2 * (size_t)S];
    float nrm = sqrtf(v0 * v0 + v1 * v1 + v2 * v2) + EPSV;
    nb[((size_t)b * C + c) * S + sm] = nrm;
    s1 += nrm; s2 += nrm * nrm;
  }
  __shared__ float r1[256], r2[256];
  r1[tid] = s1; r2[tid] = s2;
  __syncthreads();
  for (int o = 128; o > 0; o >>= 1) {
    if (tid < o) { r1[tid] += r1[tid + o]; r2[tid] += r2[tid + o]; }
    __syncthreads();
  }
  if (tid == 0) {
    atomicAdd(&stats[c], r1[0]);
    atomicAdd(&stats[MAXC + c], r2[0]);
  }
}

__global__ void k_finstats(float* stats, int C, float cnt) {
  int c = blockIdx.x * blockDim.x + threadIdx.x;
  if (c < C) {
    float mu  = stats[c] / cnt;
    float var = stats[MAXC + c] / cnt - mu * mu;
    stats[2 * MAXC + c] = mu;
    stats[3 * MAXC + c] = rsqrtf(var + BNE);
  }
}

// vn-LeakyReLU (slope=0) combine + mean over trailing k.  Nout = S/J.
__global__ void k_combine(const float* __restrict__ P, const float* __restrict__ D,
                          const float* __restrict__ nb, const float* __restrict__ stats,
                          const float* __restrict__ g, const float* __restrict__ be,
                          float* __restrict__ Y, int C, int S, int J, int Bn) {
  long tid = (long)blockIdx.x * blockDim.x + threadIdx.x;
  long Nout = S / J;
  long tot = (long)Bn * C * Nout;
  if (tid >= tot) return;
  long n = tid % Nout, c = (tid / Nout) % C, b = tid / (Nout * C);
  float mu = stats[2 * MAXC + c], rs = stats[3 * MAXC + c];
  float gg = g[c], bb = be[c];
  const float* Pb = P + ((size_t)b * C + c) * 3 * S;
  const float* Db = D + ((size_t)b * C + c) * 3 * S;
  const float* Nb = nb + ((size_t)b * C + c) * S;
  float a0 = 0.f, a1 = 0.f, a2 = 0.f;
  for (int j = 0; j < J; ++j) {
    long m = n * J + j;
    float nrm = Nb[m];
    float sc  = (gg * (nrm - mu) * rs + bb) / nrm;
    float p0 = Pb[m] * sc, p1 = Pb[(size_t)S + m] * sc, p2 = Pb[2 * (size_t)S + m] * sc;
    float d0 = Db[m], d1 = Db[(size_t)S + m], d2 = Db[2 * (size_t)S + m];
    float dot = p0 * d0 + p1 * d1 + p2 * d2;
    if (dot >= 0.f) { a0 += p0; a1 += p1; a2 += p2; }
    else {
      float f = dot / (d0 * d0 + d1 * d1 + d2 * d2 + EPSV);
      a0 += p0 - f * d0; a1 += p1 - f * d1; a2 += p2 - f * d2;
    }
  }
  float inv = 1.f / (float)J;
  float* Yb = Y + ((size_t)b * C + c) * 3 * Nout;
  Yb[n] = a0 * inv; Yb[Nout + n] = a1 * inv; Yb[2 * Nout + n] = a2 * inv;
}

// bn-only rescale (c3 stage): Y = P * nbn / norm
__global__ void k_bnscale(const float* __restrict__ P, const float* __restrict__ nb,
                          const float* __restrict__ stats, const float* __restrict__ g,
                          const float* __restrict__ be, float* __restrict__ Y,
                          int C, int S, int Bn) {
  long tid = (long)blockIdx.x * blockDim.x + threadIdx.x;
  long tot = (long)Bn * C * S;
  if (tid >= tot) return;
  long m = tid % S, c = (tid / S) % C, b = tid / ((long)S * C);
  float mu = stats[2 * MAXC + c], rs = stats[3 * MAXC + c];
  float nrm = nb[((size_t)b * C + c) * S + m];
  float sc  = (g[c] * (nrm - mu) * rs + be[c]) / nrm;
  const float* Pb = P + ((size_t)b * C + c) * 3 * S;
  float* Yb = Y + ((size_t)b * C + c) * 3 * S;
  Yb[m] = Pb[m] * sc;
  Yb[(size_t)S + m] = Pb[(size_t)S + m] * sc;
  Yb[2 * (size_t)S + m] = Pb[2 * (size_t)S + m] * sc;
}

__global__ void k_meanN(const float* __restrict__ X, float* __restrict__ Y,
                        int C, int N, int Bn) {
  int tid = blockIdx.x * blockDim.x + threadIdx.x;
  int tot = Bn * C * 3;
  if (tid >= tot) return;
  const float* p = X + (size_t)tid * N;
  float s = 0.f;
  for (int n = 0; n < N; ++n) s += p[n];
  Y[tid] = s / (float)N;
}

__global__ void k_concat(const float* __restrict__ H, const float* __restrict__ xg,
                         float* __restrict__ Y, int C1, int N, int Bn) {
  long tid = (long)blockIdx.x * blockDim.x + threadIdx.x;
  int Ct = 2 * C1;
  long tot = (long)Bn * Ct * 3 * N;
  if (tid >= tot) return;
  long n = tid % N, dd = (tid / N) % 3, c = (tid / ((long)3 * N)) % Ct,
       b = tid / ((long)3 * N * Ct);
  float v;
  if (c < C1) v = H[(((size_t)b * C1 + c) * 3 + dd) * N + n];
  else        v = xg[((size_t)b * C1 + (c - C1)) * 3 + dd];
  Y[(((size_t)b * Ct + c) * 3 + dd) * N + n] = v;
}

// out[b,n,dd] = sum_c W[c] * H[b,c,dd,n]   (final 341->1 + transpose)
__global__ void k_out(const float* __restrict__ H, const float* __restrict__ W,
                      float* __restrict__ out, int C, int N, int Bn) {
  int tid = blockIdx.x * blockDim.x + threadIdx.x;
  int tot = Bn * N * 3;
  if (tid >= tot) return;
  int dd = tid % 3, n = (tid / 3) % N, b = tid / (3 * N);
  float s = 0.f;
  for (int c = 0; c < C; ++c)
    s += W[c] * H[(((size_t)b * C + c) * 3 + dd) * N + n];
  out[((size_t)b * N + n) * 3 + dd] = s;
}

// ---------------------------------------------------------------------------
extern "C" void kernel_launch(void* const* d_in, const int* in_sizes, int n_in,
                              void* d_out, int out_size, void* d_ws, size_t ws_size,
                              hipStream_t stream) {
  (void)n_in; (void)out_size; (void)ws_size;
  const int Bq = in_sizes[1];
  const int N  = in_sizes[0] / (3 * Bq);
  const int Kk = 10;

  const float* x = (const float*)d_in[0];
  const float* t = (const float*)d_in[1];
  const float *teWf[3] = {(const float*)d_in[2], (const float*)d_in[5], (const float*)d_in[8]};
  const float *teW [3] = {(const float*)d_in[3], (const float*)d_in[6], (const float*)d_in[9]};
  const float *teB [3] = {(const float*)d_in[4], (const float*)d_in[7], (const float*)d_in[10]};
  const float* cpWf = (const float*)d_in[11]; const float* cpWd = (const float*)d_in[12];
  const float* cpg  = (const float*)d_in[13]; const float* cpb  = (const float*)d_in[14];
  const float* c1Wf = (const float*)d_in[15]; const float* c1Wd = (const float*)d_in[16];
  const float* c1g  = (const float*)d_in[17]; const float* c1b  = (const float*)d_in[18];
  const float* f1Wf = (const float*)d_in[19]; const float* f1Wd = (const float*)d_in[20];
  const float* f1g  = (const float*)d_in[21]; const float* f1b  = (const float*)d_in[22];
  const float* f2Wf = (const float*)d_in[23]; const float* f2Wd = (const float*)d_in[24];
  const float* f2g  = (const float*)d_in[25]; const float* f2b  = (const float*)d_in[26];
  const float* f3Wf = (const float*)d_in[27]; const float* f3Wd = (const float*)d_in[28];
  const float* f3g  = (const float*)d_in[29]; const float* f3b  = (const float*)d_in[30];
  const float* fc1Wf = (const float*)d_in[31]; const float* fc1Wd = (const float*)d_in[32];
  const float* fc1g  = (const float*)d_in[33]; const float* fc1b  = (const float*)d_in[34];
  const float* fc2Wf = (const float*)d_in[35]; const float* fc2Wd = (const float*)d_in[36];
  const float* fc2g  = (const float*)d_in[37]; const float* fc2b  = (const float*)d_in[38];
  const float* fc3W  = (const float*)d_in[39];
  const float* c2Wf  = (const float*)d_in[40]; const float* c2Wd = (const float*)d_in[41];
  const float* c2g   = (const float*)d_in[42]; const float* c2b  = (const float*)d_in[43];
  const float* c3W   = (const float*)d_in[44];
  const float* bn3g  = (const float*)d_in[45]; const float* bn3b = (const float*)d_in[46];
  const float* outW  = (const float*)d_in[47];

  // workspace layout (floats)
  float* wsf = (float*)d_ws;
  float* TE0   = wsf + 0;
  float* TE1   = wsf + 256;
  float* TE2   = wsf + 512;
  float* XX    = wsf + 1024;                  // B*N
  float* STATS = wsf + 1024 + 16384;          // 4*MAXC
  float* S0    = STATS + 4 * MAXC;            // B*341*3
  float* S1    = S0 + 8192;                   // B*170*3
  float* XG    = S1 + 4096;                   // B*21*3
  int*   IDX   = (int*)(wsf + 40960);         // B*N*Kk
  const size_t BIGSZ = 17u * 1024u * 1024u;   // 17M floats per region
  float* R0 = wsf + 262144;
  float* R1 = R0 + BIGSZ;
  float* R2 = R1 + BIGSZ;   // P scratch
  float* R3 = R2 + BIGSZ;   // D scratch / bn out
  float* R4 = R3 + BIGSZ;   // norm scratch
  float* R5 = R4 + BIGSZ;   // h1 (kept live for concat)

  auto cdiv = [](long a, long b) { return (int)((a + b - 1) / b); };

  auto gemm = [&](const float* W, const float* X, float* Y, const float* te,
                  int Co, int Ci, int Mc) {
    long waves = (long)Bq * cdiv(Co, 16) * cdiv(Mc, 16);
    k_gemm<<<cdiv(waves, 4), 128, 0, stream>>>(W, X, Y, te, Bq, Co, Ci, Mc);
  };

  // full vn_lrelu block: X(B,Ci,3,S) -> Y(B,Co,3,S/J)
  auto vnlrelu = [&](const float* Xin, float* Yout, int Co, int Ci, int S, int J,
                     const float* Wf, const float* Wd, const float* g,
                     const float* be, const float* te) {
    int Mc = 3 * S;
    gemm(Wf, Xin, R2, te, Co, Ci, Mc);
    gemm(Wd, Xin, R3, nullptr, Co, Ci, Mc);
    k_zero<<<cdiv(4 * MAXC, 256), 256, 0, stream>>>(STATS, 4 * MAXC);
    k_normstats<<<dim3(32, Co), 256, 0, stream>>>(R2, R4, STATS, Co, S, Bq);
    k_finstats<<<cdiv(MAXC, 256), 256, 0, stream>>>(STATS, Co, (float)((long)Bq * S));
    long tot = (long)Bq * Co * (S / J);
    k_combine<<<cdiv(tot, 256), 256, 0, stream>>>(R2, R3, R4, STATS, g, be, Yout,
                                                  Co, S, J, Bq);
  };

  // 0) time embeddings
  k_te<<<Bq, 64, 0, stream>>>(t, teWf[0], teW[0], teB[0], TE0, 21);
  k_te<<<Bq, 64, 0, stream>>>(t, teWf[1], teW[1], teB[1], TE1, 21);
  k_te<<<Bq, 64, 0, stream>>>(t, teWf[2], teW[2], teB[2], TE2, 42);

  // 1) kNN graph (WMMA Gram tiles)
  k_xx<<<cdiv(Bq * N, 256), 256, 0, stream>>>(x, XX, Bq * N);
  k_knn<<<Bq * (N / 16), 32, 0, stream>>>(x, XX, IDX, N);

  // 2) edge features  feat(B,3,3,N,k) -> R0
  long ftot = (long)Bq * N * Kk;
  k_feat<<<cdiv(ftot, 256), 256, 0, stream>>>(x, IDX, R0, N, Kk, (int)ftot);

  // 3) cp: (3 -> 21) over (N,k), mean over k  -> R1 (B,21,3,N)
  vnlrelu(R0, R1, 21, 3, N * Kk, Kk, cpWf, cpWd, cpg, cpb, nullptr);
  // 4) c1 (te0): 21 -> 21  -> R5 (kept for concat)
  vnlrelu(R1, R5, 21, 21, N, 1, c1Wf, c1Wd, c1g, c1b, TE0);

  // 5) STN
  vnlrelu(R5, R0, 21, 21, N, 1, f1Wf, f1Wd, f1g, f1b, TE1);   // f1 (te1)
  vnlrelu(R0, R1, 42, 21, N, 1, f2Wf, f2Wd, f2g, f2b, nullptr); // f2
  vnlrelu(R1, R0, 341, 42, N, 1, f3Wf, f3Wd, f3g, f3b, nullptr); // f3
  k_meanN<<<cdiv(Bq * 341 * 3, 256), 256, 0, stream>>>(R0, S0, 341, N, Bq);
  vnlrelu(S0, S1, 170, 341, 1, 1, fc1Wf, fc1Wd, fc1g, fc1b, nullptr); // fc1
  vnlrelu(S1, S0, 85, 170, 1, 1, fc2Wf, fc2Wd, fc2g, fc2b, nullptr);  // fc2
  gemm(fc3W, S0, XG, nullptr, 21, 85, 3);                             // fc3

  // 6) concat [h1 ; broadcast(xg)] -> R1 (B,42,3,N)
  long ctot = (long)Bq * 42 * 3 * N;
  k_concat<<<cdiv(ctot, 256), 256, 0, stream>>>(R5, XG, R1, 21, N, Bq);

  // 7) c2 (te2): 42 -> 42  -> R0
  vnlrelu(R1, R0, 42, 42, N, 1, c2Wf, c2Wd, c2g, c2b, TE2);

  // 8) c3: vnlin 42->341 then vnbn only  -> R3
  gemm(c3W, R0, R1, nullptr, 341, 42, 3 * N);
  k_zero<<<cdiv(4 * MAXC, 256), 256, 0, stream>>>(STATS, 4 * MAXC);
  k_normstats<<<dim3(32, 341), 256, 0, stream>>>(R1, R4, STATS, 341, N, Bq);
  k_finstats<<<cdiv(MAXC, 256), 256, 0, stream>>>(STATS, 341, (float)((long)Bq * N));
  long btot = (long)Bq * 341 * N;
  k_bnscale<<<cdiv(btot, 256), 256, 0, stream>>>(R1, R4, STATS, bn3g, bn3b, R3,
                                                 341, N, Bq);

  // 9) out_W projection + transpose -> d_out (B,N,3)
  k_out<<<cdiv((long)Bq * N * 3, 256), 256, 0, stream>>>(R3, outW, (float*)d_out,
                                                         341, N, Bq);
}